// KGReasoning_3212635537979
// MI455X (gfx1250) — compile-verified
//
#include <hip/hip_runtime.h>

// ---------------------------------------------------------------------------
// max-product "matvec" with argmax over 8192x8192 f32 (bandwidth-bound).
//   new_emb[t] = max_h e[h]*rel[h,t];  r_argmax[t] = first argmax_h
// rel is 256 MB streamed exactly once -> ~11us floor @ 23.3 TB/s. WMMA (sum
// semiring) cannot express a max/argmax reduction, and at 0.52 FLOP/byte the
// problem is pure memory-path anyway.
// Cross-slice merge via packed u64 global atomic max:
//   packed = (f32bits(val) << 32) | ~h
// Nonneg f32 bits are u32-monotone; ~h makes equal values prefer smallest h
// (first-max-wins, matching the reference), and is order-independent =>
// deterministic under atomics.
// ---------------------------------------------------------------------------

#define N            8192
#define SLICES       32
#define ROWS         (N / SLICES)        // 256 rows of rel per slice
#define TPB          128                 // 4 wave32 per block
#define CPT          4                   // columns per thread (one b128 load)
#define COLS_PER_BLK (TPB * CPT)         // 512
#define COL_TILES    (N / COLS_PER_BLK)  // 16
#define PF           8                   // prefetch distance (rows)

typedef float v4f __attribute__((ext_vector_type(4)));

__global__ __launch_bounds__(256) void kg_init_ws(unsigned long long* __restrict__ ws) {
    int t = blockIdx.x * blockDim.x + threadIdx.x;
    if (t < N) ws[t] = 0ull;
}

__global__ __launch_bounds__(TPB) void kg_maxprod(const float* __restrict__ emb,
                                                  const float* __restrict__ rel,
                                                  unsigned long long* __restrict__ ws) {
    __shared__ float s_emb[ROWS];

    const int h0 = blockIdx.y * ROWS;

    // Stage this slice's embedding values into LDS once (broadcast reads after).
    for (int i = threadIdx.x; i < ROWS; i += TPB)
        s_emb[i] = emb[h0 + i];
    __syncthreads();

    const int col = blockIdx.x * COLS_PER_BLK + threadIdx.x * CPT;
    const float* __restrict__ p = rel + (size_t)h0 * N + col;

    float m0 = 0.0f, m1 = 0.0f, m2 = 0.0f, m3 = 0.0f;
    int   i0 = 0,    i1 = 0,    i2 = 0,    i3 = 0;

#define KG_BODY(DO_PF)                                                        \
    {                                                                         \
        v4f r = __builtin_nontemporal_load((const v4f*)p);                    \
        if (DO_PF)                                                            \
            __builtin_prefetch((const void*)(p + PF * (size_t)N), 0, 0);      \
        const float e = s_emb[hh];          /* LDS broadcast read */          \
        const int   h = h0 + hh;                                              \
        float v0 = e * r[0];                                                  \
        float v1 = e * r[1];                                                  \
        float v2 = e * r[2];                                                  \
        float v3 = e * r[3];                                                  \
        if (v0 > m0) { m0 = v0; i0 = h; }   /* strict '>' keeps earliest h */ \
        if (v1 > m1) { m1 = v1; i1 = h; }                                     \
        if (v2 > m2) { m2 = v2; i2 = h; }                                     \
        if (v3 > m3) { m3 = v3; i3 = h; }                                     \
        p += N;                                                               \
    }

    // Main loop: unconditional in-bounds prefetch, 8 loads in flight per wave.
#pragma unroll 8
    for (int hh = 0; hh < ROWS - PF; ++hh)
        KG_BODY(1)
    // Tail: last PF rows, no prefetch (keeps all prefetches in-bounds).
#pragma unroll
    for (int hh = ROWS - PF; hh < ROWS; ++hh)
        KG_BODY(0)
#undef KG_BODY

    // Merge this slice's partial (max, argmax) with one u64 atomic per column.
    unsigned long long P0 = ((unsigned long long)__float_as_uint(m0) << 32) | (unsigned)(~i0);
    unsigned long long P1 = ((unsigned long long)__float_as_uint(m1) << 32) | (unsigned)(~i1);
    unsigned long long P2 = ((unsigned long long)__float_as_uint(m2) << 32) | (unsigned)(~i2);
    unsigned long long P3 = ((unsigned long long)__float_as_uint(m3) << 32) | (unsigned)(~i3);
    atomicMax(&ws[col + 0], P0);
    atomicMax(&ws[col + 1], P1);
    atomicMax(&ws[col + 2], P2);
    atomicMax(&ws[col + 3], P3);
}

__global__ __launch_bounds__(256) void kg_finalize(const unsigned long long* __restrict__ ws,
                                                   float* __restrict__ out) {
    int t = blockIdx.x * blockDim.x + threadIdx.x;
    if (t < N) {
        unsigned long long pk = ws[t];
        out[t]     = __uint_as_float((unsigned)(pk >> 32));   // max value
        out[N + t] = (float)(~(unsigned)pk);                  // argmax index
    }
}

extern "C" void kernel_launch(void* const* d_in, const int* in_sizes, int n_in,
                              void* d_out, int out_size, void* d_ws, size_t ws_size,
                              hipStream_t stream) {
    (void)in_sizes; (void)n_in; (void)out_size; (void)ws_size;

    const float* emb = (const float*)d_in[0];   // (1, 8192) f32
    const float* rel = (const float*)d_in[1];   // (8192, 8192) f32, row-major [h][t]
    float* out = (float*)d_out;                 // 16384 f32: [vals | argmax-as-float]
    unsigned long long* ws = (unsigned long long*)d_ws;  // 8192 * 8B = 64 KB

    // 1) zero the packed accumulators (must be per-call for graph replay)
    kg_init_ws<<<N / 256, 256, 0, stream>>>(ws);

    // 2) streaming max-product: 16 column tiles x 32 h-slices = 512 blocks
    dim3 grid(COL_TILES, SLICES);
    kg_maxprod<<<grid, TPB, 0, stream>>>(emb, rel, ws);

    // 3) unpack into d_out
    kg_finalize<<<N / 256, 256, 0, stream>>>(ws, out);
}